// GCNRegressor_7997229105851
// MI455X (gfx1250) — compile-verified
//
#include <hip/hip_runtime.h>
#include <hip/hip_bf16.h>

// ---------------------------------------------------------------------------
// GCN 2-layer regressor for MI455X (gfx1250, wave32).
//   L2 (192MB) holds h/agg1 (51MB each) -> edge scatter is L2-resident.
//   GEMM1 uses V_WMMA_F32_16X16X4_F32 (fp32 in/out, matches reference dtype).
// ---------------------------------------------------------------------------

typedef __attribute__((ext_vector_type(2))) float v2f;
typedef __attribute__((ext_vector_type(8))) float v8f;

#define IN_CH 128   // = HID_CH in this model

// --- degree / norm ----------------------------------------------------------
__global__ void k_deg_init(float* __restrict__ deg, int n) {
    int i = blockIdx.x * blockDim.x + threadIdx.x;
    if (i < n) deg[i] = 1.0f;                      // self-loop
}

__global__ void k_deg_count(const int* __restrict__ dst, float* __restrict__ deg, int e) {
    int i = blockIdx.x * blockDim.x + threadIdx.x;
    if (i < e) atomicAdd(&deg[dst[i]], 1.0f);
}

__global__ void k_dinv(const float* __restrict__ deg, float* __restrict__ dinv, int n) {
    int i = blockIdx.x * blockDim.x + threadIdx.x;
    if (i < n) dinv[i] = rsqrtf(deg[i]);           // deg >= 1 always
}

// --- layer 1 GEMM: h = x @ W1 via fp32 WMMA, epilogue seeds self-loop term --
// One wave computes a 16x128 row-block: 8 accumulators of 16x16, K-loop of 32
// V_WMMA_F32_16X16X4_F32 steps. W1 (64KB) staged in LDS per block.
//
// Fragment layouts (wave32, per ISA 7.12.2):
//   A 16x4 f32 : lanes 0-15 -> row M=lane, v0=K0 v1=K1 ; lanes 16-31 -> K2,K3
//   B  4x16 f32: lanes 0-15 -> col N=lane, v0=K0 v1=K1 ; lanes 16-31 -> K2,K3
//   C/D 16x16  : VGPR j -> M=j (lanes 0-15) / M=j+8 (lanes 16-31), N=lane&15
__global__ __launch_bounds__(256) void k_gemm1_wmma(
    const float* __restrict__ x, const float* __restrict__ W1,
    const float* __restrict__ dinv,
    float* __restrict__ h, float* __restrict__ agg1,
    int n_nodes, int mtiles) {
    __shared__ float lw[IN_CH * IN_CH];            // 64 KB
    const int tid = threadIdx.x;
    for (int i = tid; i < IN_CH * IN_CH; i += 256) lw[i] = W1[i];
    __syncthreads();

    const int wid  = tid >> 5;
    const int lane = tid & 31;
    const int half = lane >> 4;                    // 0: K pair {0,1}, 1: {2,3}
    const int l    = lane & 15;
    const int tile = blockIdx.x * 8 + wid;
    if (tile >= mtiles) return;                    // wave-uniform exit
    const int m0 = tile * 16;

    v8f acc[8] = {};                               // 8 N-tiles x 8 VGPRs

    const int arow = min(m0 + l, n_nodes - 1);     // clamp (n%16==0 normally)
    const float* ap = x + (size_t)arow * IN_CH;

    for (int kk = 0; kk < IN_CH; kk += 4) {
        const int ka = kk + 2 * half;
        v2f a;
        a.x = ap[ka];
        a.y = ap[ka + 1];
#pragma unroll
        for (int nt = 0; nt < 8; ++nt) {
            const int nc = nt * 16 + l;
            v2f b;
            b.x = lw[ka * IN_CH + nc];
            b.y = lw[(ka + 1) * IN_CH + nc];
            acc[nt] = __builtin_amdgcn_wmma_f32_16x16x4_f32(
                false, a, false, b, (short)0, acc[nt], false, false);
        }
    }

    // Epilogue: store h and agg1 = h * dinv^2 (self-loop contribution).
#pragma unroll
    for (int j = 0; j < 8; ++j) {
        const int row = m0 + j + 8 * half;
        if (row >= n_nodes) continue;
        const float di = dinv[row];
        const float sc = di * di;
#pragma unroll
        for (int nt = 0; nt < 8; ++nt) {
            const float c = acc[nt][j];
            const size_t idx = (size_t)row * IN_CH + nt * 16 + l;
            h[idx]    = c;
            agg1[idx] = c * sc;
        }
    }
}

// --- layer 1 edge scatter: agg1[dst] += h[src] * dinv[src]*dinv[dst] --------
// One wave per edge: lane handles 4 channels (float4 gather, 4 f32 atomics).
__global__ __launch_bounds__(256) void k_edge_agg1(
    const int* __restrict__ src, const int* __restrict__ dst,
    const float* __restrict__ dinv, const float* __restrict__ h,
    float* __restrict__ agg1, int e) {
    const int w    = blockIdx.x * 8 + (threadIdx.x >> 5);
    const int lane = threadIdx.x & 31;
    if (w >= e) return;
    const int s = src[w], d = dst[w];
    const float norm = dinv[s] * dinv[d];
    const float4 hv = ((const float4*)(h + (size_t)s * IN_CH))[lane];
    float* op = agg1 + (size_t)d * IN_CH + lane * 4;
    atomicAdd(op + 0, hv.x * norm);
    atomicAdd(op + 1, hv.y * norm);
    atomicAdd(op + 2, hv.z * norm);
    atomicAdd(op + 3, hv.w * norm);
}

// --- layer 2 fused: h2 = relu(agg1+b1) @ W2 ; out = h2*dinv^2 + b2 ----------
// One wave per node; lane dot over 4 channels, shuffle tree reduction.
__global__ __launch_bounds__(256) void k_gemv2(
    const float* __restrict__ agg1, const float* __restrict__ b1,
    const float* __restrict__ W2, const float* __restrict__ b2,
    const float* __restrict__ dinv,
    float* __restrict__ h2, float* __restrict__ out, int n) {
    const int node = blockIdx.x * 8 + (threadIdx.x >> 5);
    const int lane = threadIdx.x & 31;
    if (node >= n) return;
    const float4 a  = ((const float4*)(agg1 + (size_t)node * IN_CH))[lane];
    const float4 bb = ((const float4*)b1)[lane];
    const float4 w  = ((const float4*)W2)[lane];
    float v = fmaxf(a.x + bb.x, 0.f) * w.x + fmaxf(a.y + bb.y, 0.f) * w.y +
              fmaxf(a.z + bb.z, 0.f) * w.z + fmaxf(a.w + bb.w, 0.f) * w.w;
#pragma unroll
    for (int off = 16; off > 0; off >>= 1) v += __shfl_xor(v, off, 32);
    if (lane == 0) {
        const float di = dinv[node];
        h2[node]  = v;
        out[node] = v * di * di + b2[0];           // self-loop + bias init
    }
}

// --- layer 2 edge scatter: out[dst] += h2[src] * dinv[src]*dinv[dst] --------
__global__ void k_edge_agg2(const int* __restrict__ src, const int* __restrict__ dst,
                            const float* __restrict__ dinv, const float* __restrict__ h2,
                            float* __restrict__ out, int e) {
    int i = blockIdx.x * blockDim.x + threadIdx.x;
    if (i < e) {
        const int s = src[i], d = dst[i];
        atomicAdd(&out[d], h2[s] * dinv[s] * dinv[d]);
    }
}

extern "C" void kernel_launch(void* const* d_in, const int* in_sizes, int n_in,
                              void* d_out, int out_size, void* d_ws, size_t ws_size,
                              hipStream_t stream) {
    const float* x  = (const float*)d_in[0];
    const int*   ei = (const int*)d_in[1];
    const float* W1 = (const float*)d_in[2];
    const float* b1 = (const float*)d_in[3];
    const float* W2 = (const float*)d_in[4];
    const float* b2 = (const float*)d_in[5];

    const int n = in_sizes[0] / IN_CH;   // 100000
    const int e = in_sizes[1] / 2;       // 1600000
    const int* src = ei;
    const int* dst = ei + e;

    // Workspace layout (floats): deg[n] | dinv[n] | h[n*128] | agg1[n*128] | h2[n]
    float* ws   = (float*)d_ws;
    float* deg  = ws;
    float* dinv = deg + n;
    float* h    = dinv + n;
    float* agg1 = h + (size_t)n * IN_CH;
    float* h2   = agg1 + (size_t)n * IN_CH;
    float* out  = (float*)d_out;

    const int mtiles = (n + 15) / 16;

    k_deg_init <<<(n + 255) / 256, 256, 0, stream>>>(deg, n);
    k_deg_count<<<(e + 255) / 256, 256, 0, stream>>>(dst, deg, e);
    k_dinv     <<<(n + 255) / 256, 256, 0, stream>>>(deg, dinv, n);
    k_gemm1_wmma<<<(mtiles + 7) / 8, 256, 0, stream>>>(x, W1, dinv, h, agg1, n, mtiles);
    k_edge_agg1<<<(e + 7) / 8, 256, 0, stream>>>(src, dst, dinv, h, agg1, e);
    k_gemv2    <<<(n + 7) / 8, 256, 0, stream>>>(agg1, b1, W2, b2, dinv, h2, out, n);
    k_edge_agg2<<<(e + 255) / 256, 256, 0, stream>>>(src, dst, dinv, h2, out, e);
}